// Stackgram_4337916969520
// MI455X (gfx1250) — compile-verified
//
#include <hip/hip_runtime.h>

// Stackgram: out[b,t,h,w] = valid ? x[b,0,idx(t,h,w),t] : 0
// B=2, S=512, T=180.  Pure gather + streaming store: HBM-store-bound (~377MB).
// Strategy: per (b,t) stage the 512-float line into LDS with CDNA5 async
// global->LDS copies, gather via ds_load, write float4 non-temporal stores.

constexpr int B_ = 2;
constexpr int S_ = 512;
constexpr int T_ = 180;
constexpr int ROWS = 16;   // output rows per block

typedef float v4f __attribute__((ext_vector_type(4)));

__global__ __launch_bounds__(256) void stackgram_kernel(
    const float* __restrict__ x, float* __restrict__ out)
{
    __shared__ float line[S_];

    const int bt = blockIdx.x;        // == b*T + t (matches output layout)
    const int t  = bt % T_;
    const int b  = bt / T_;

    // ---- Stage x[b, 0, 0:S, t] into LDS (stride T floats in memory) ----
    // CDNA5 async copy: global_load_async_to_lds_b32 vdst(lds addr), vaddr(u32 off), saddr
    {
        const unsigned long long gbase =
            (unsigned long long)(uintptr_t)x +
            ((unsigned long long)(b * S_ * T_ + t)) * 4ull;
        for (int s = threadIdx.x; s < S_; s += 256) {
            unsigned lds_off = (unsigned)(uintptr_t)(&line[s]);  // LDS byte offset
            unsigned voff    = (unsigned)(s * T_) * 4u;          // global byte offset
            asm volatile("global_load_async_to_lds_b32 %0, %1, %2"
                         :: "v"(lds_off), "v"(voff), "s"(gbase) : "memory");
        }
        asm volatile("s_wait_asynccnt 0" ::: "memory");
    }
    __syncthreads();

    // ---- Index math (mirrors reference float32 op order; no FMA contraction) ----
    const float step = 2.0f / 511.0f;                    // linspace(-1,1,512) step
    const float th   = (float)t * 1.7453292519943295e-2f; // float32(pi/180)
    const float sn = sinf(th);
    const float cs = cosf(th);

    const int lane   = threadIdx.x & 127;   // float4 slot within a row
    const int rowoff = threadIdx.x >> 7;    // 0 or 1: two rows per iteration
    const int w0     = lane * 4;

    // precompute bw terms for the 4 consecutive w handled by this thread
    float bw[4];
#pragma unroll
    for (int j = 0; j < 4; ++j)
        bw[j] = __fadd_rn(__fmul_rn((float)(w0 + j), step), -1.0f);

    const int hbase = blockIdx.y * ROWS;

    for (int r = rowoff; r < ROWS; r += 2) {
        const int h = hbase + r;
        const float bh   = __fadd_rn(__fmul_rn((float)h, step), -1.0f);
        const float snbh = __fmul_rn(sn, bh);

        v4f vec;
#pragma unroll
        for (int j = 0; j < 4; ++j) {
            float gy  = __fsub_rn(__fmul_rn(cs, bw[j]), snbh);
            float pix = __fmul_rn(__fmul_rn(__fadd_rn(gy, 1.0f), 0.5f), 511.0f);
            int   idx = (int)rintf(pix);            // v_rndne: round-half-even
            int   ci  = idx < 0 ? 0 : (idx > S_ - 1 ? S_ - 1 : idx);
            float val = line[ci];                   // ds_load_b32 gather
            if ((unsigned)idx > (unsigned)(S_ - 1)) val = 0.0f;
            vec[j] = val;
        }

        // out[((b*T + t)*S + h)*S + w0 .. +3], streaming 128-bit NT store
        size_t o = (((size_t)bt * S_ + (size_t)h) * S_) + (size_t)w0;
        __builtin_nontemporal_store(vec, (v4f*)(out + o));
    }
}

extern "C" void kernel_launch(void* const* d_in, const int* in_sizes, int n_in,
                              void* d_out, int out_size, void* d_ws, size_t ws_size,
                              hipStream_t stream) {
    const float* x = (const float*)d_in[0];
    float* out = (float*)d_out;
    (void)in_sizes; (void)n_in; (void)out_size; (void)d_ws; (void)ws_size;

    dim3 grid(B_ * T_, S_ / ROWS);   // (360, 32) blocks
    dim3 block(256);                 // 8 wave32s
    stackgram_kernel<<<grid, block, 0, stream>>>(x, out);
}